// VectorQuantizer_4415226380350
// MI455X (gfx1250) — compile-verified
//
#include <hip/hip_runtime.h>
#include <hip/hip_bf16.h>
#include <math.h>

typedef __attribute__((ext_vector_type(2))) float v2f;
typedef __attribute__((ext_vector_type(8))) float v8f;

// Problem constants
#define NROWS   131072      // 32*64*64 flattened NHWC rows
#define KCODES  512
#define DDIM    64
#define HWSZ    4096        // 64*64
#define BSTRIDE 262144      // 64*64*64 per batch in NCHW

// d_out layout (floats), in reference return order
#define OFF_DIST   0u
#define OFF_QUANT  67108864u
#define OFF_LOSS   75497472u
#define OFF_ENC    75497473u
#define OFF_IDX    142606337u
#define OFF_PERP   142737409u

// ---------------------------------------------------------------------------
// Kernel 0: zero counts/loss + precompute wsq[512] = ||w_k||^2
// (one block, 512 threads)
// ---------------------------------------------------------------------------
__global__ void vq_init_kernel(const float* __restrict__ w,
                               int* __restrict__ counts,
                               float* __restrict__ loss_acc,
                               float* __restrict__ wsq_tab) {
    const int t = threadIdx.x;
    counts[t] = 0;
    if (t == 0) *loss_acc = 0.0f;
    const float4* wr = (const float4*)(w + t * DDIM);
    float s = 0.0f;
    #pragma unroll
    for (int i = 0; i < 16; ++i) {
        float4 v = wr[i];
        s += v.x * v.x + v.y * v.y + v.z * v.z + v.w * v.w;
    }
    wsq_tab[t] = s;
}

// ---------------------------------------------------------------------------
// Kernel 1: fused distances GEMM (fp32 WMMA) + argmin + one-hot + indices
//   D = (-2*A) x B + C,  C preloaded with ||x||^2 + ||w||^2
// One wave handles 16 rows x 512 codes. 8 waves per block.
// ---------------------------------------------------------------------------
__global__ __launch_bounds__(256) void vq_dist_kernel(
    const float* __restrict__ x,        // NCHW [32,64,64,64]
    const float* __restrict__ w,        // [512,64] row-major
    const float* __restrict__ wsq_tab,  // [512] precomputed ||w||^2
    float* __restrict__ dist,           // [N,512]
    float* __restrict__ enc,            // [N,512] one-hot
    float* __restrict__ enc_idx,        // [N] (indices stored as float)
    int*   __restrict__ counts,         // [512]
    float* __restrict__ loss_acc)       // scalar accumulator
{
    const int lane = threadIdx.x & 31;
    const int wave = threadIdx.x >> 5;
    const int wid  = blockIdx.x * 8 + wave;
    const int r0   = wid * 16;              // first of this wave's 16 rows
    const int l    = lane & 15;             // column / row-in-tile index
    const int h    = lane >> 4;             // lane half

    // ---- Load A fragments (16x64 x-tile), fold -2, compute ||x||^2 --------
    // flat[m][d] = x[b*BSTRIDE + d*HWSZ + hw], m = b*HWSZ + hw
    const int m_a  = r0 + l;
    const int b_a  = m_a >> 12;
    const int hw_a = m_a & 4095;
    const float* xbase = x + b_a * BSTRIDE + hw_a;

    float af[32];
    float psum = 0.0f;
    #pragma unroll
    for (int s = 0; s < 16; ++s) {
        float a0 = xbase[(size_t)(4 * s + 2 * h) * HWSZ];
        float a1 = xbase[(size_t)(4 * s + 2 * h + 1) * HWSZ];
        psum += a0 * a0 + a1 * a1;
        af[2 * s]     = -2.0f * a0;
        af[2 * s + 1] = -2.0f * a1;
    }
    // lane l (either half) now holds ||x||^2 of row (r0+l)   (one-time shuffle)
    float xsq = psum + __shfl_xor(psum, 16);
    // broadcast: xsqb[r] = ||x||^2 of row (r + 8*h) for this lane's half
    float xsqb[8];
    #pragma unroll
    for (int r = 0; r < 8; ++r)
        xsqb[r] = __shfl(xsq, r + 8 * h);

    float minv[8];
    int   mini[8];
    #pragma unroll
    for (int r = 0; r < 8; ++r) { minv[r] = 3.0e38f; mini[r] = 0; }

    // ---- Loop over 32 column tiles of 16 codes --------------------------
    for (int t = 0; t < 32; ++t) {
        const int n0 = t * 16;
        const float* wrow = w + (size_t)(n0 + l) * DDIM;

        // B fragments: lane (h,l) holds w[n0+l][4s+2h], w[n0+l][4s+2h+1]
        float bf[32];
        #pragma unroll
        for (int s = 0; s < 16; ++s) {
            bf[2 * s]     = wrow[4 * s + 2 * h];
            bf[2 * s + 1] = wrow[4 * s + 2 * h + 1];
        }
        const float wsq = wsq_tab[n0 + l];   // no cross-lane reduce in loop

        v8f c;
        #pragma unroll
        for (int r = 0; r < 8; ++r) c[r] = xsqb[r] + wsq;

        #pragma unroll
        for (int s = 0; s < 16; ++s) {
            v2f a; a.x = af[2 * s]; a.y = af[2 * s + 1];
            v2f bb; bb.x = bf[2 * s]; bb.y = bf[2 * s + 1];
            c = __builtin_amdgcn_wmma_f32_16x16x4_f32(
                    false, a, false, bb, (short)0, c, false, false);
        }

        // store tile + running argmin (strict '<' keeps earliest index
        // along this lane's increasing-index visit order)
        #pragma unroll
        for (int r = 0; r < 8; ++r) {
            const int row = r0 + r + 8 * h;
            dist[(size_t)row * KCODES + n0 + l] = c[r];
            if (c[r] < minv[r]) {
                minv[r] = c[r];
                mini[r] = n0 + l;
            }
        }
    }

    // ---- Butterfly min-reduce within each 16-lane half (tie -> low index)
    #pragma unroll
    for (int r = 0; r < 8; ++r) {
        float v = minv[r];
        int   i = mini[r];
        #pragma unroll
        for (int off = 8; off >= 1; off >>= 1) {
            float ov = __shfl_xor(v, off);
            int   oi = __shfl_xor(i, off);
            if (ov < v || (ov == v && oi < i)) { v = ov; i = oi; }
        }
        minv[r] = v;   // row (r + 8*half) min distance, uniform in half
        mini[r] = i;   // row (r + 8*half) argmin, uniform in half
    }

    // ---- indices / counts / loss partial (sum of min dists == sum sq err)
    if (l == 0) {
        float lsum = 0.0f;
        #pragma unroll
        for (int r = 0; r < 8; ++r) {
            const int row = r0 + r + 8 * h;
            enc_idx[row] = (float)mini[r];
            atomicAdd(&counts[mini[r]], 1);
            lsum += minv[r];
        }
        atomicAdd(loss_acc, lsum);
    }

    // ---- one-hot encoding rows ------------------------------------------
    #pragma unroll
    for (int rr = 0; rr < 16; ++rr) {
        const int idx = __shfl(mini[rr & 7], (rr >> 3) * 16);
        float* erow = enc + (size_t)(r0 + rr) * KCODES;
        #pragma unroll 4
        for (int j = 0; j < 16; ++j) {
            const int col = j * 32 + lane;
            erow[col] = (col == idx) ? 1.0f : 0.0f;
        }
    }
}

// ---------------------------------------------------------------------------
// Kernel 2: quantized_st (== quantized numerically), NCHW coalesced writer
// ---------------------------------------------------------------------------
__global__ __launch_bounds__(256) void vq_quant_kernel(
    const float* __restrict__ w,
    const float* __restrict__ enc_idx,   // float-encoded indices
    float* __restrict__ qout)            // NCHW [32,64,64,64]
{
    const int tid = blockIdx.x * 256 + threadIdx.x;  // 0 .. 8388607
    const int b   = tid >> 18;          // / 262144
    const int rem = tid & 262143;
    const int d   = rem >> 12;          // / 4096
    const int hw  = rem & 4095;
    const int m   = (b << 12) + hw;
    const int idx = (int)enc_idx[m];
    qout[tid] = w[idx * DDIM + d];
}

// ---------------------------------------------------------------------------
// Kernel 3: finalize loss + perplexity (1 block, 512 threads)
// ---------------------------------------------------------------------------
__global__ void vq_finalize_kernel(const int* __restrict__ counts,
                                   const float* __restrict__ loss_acc,
                                   float* __restrict__ out_loss,
                                   float* __restrict__ out_perp)
{
    __shared__ float sh[512];
    const int t = threadIdx.x;
    const float p = (float)counts[t] * (1.0f / (float)NROWS);
    sh[t] = p * logf(p + 1e-10f);
    __syncthreads();
    for (int s = 256; s > 0; s >>= 1) {
        if (t < s) sh[t] += sh[t + s];
        __syncthreads();
    }
    if (t == 0) {
        *out_perp = expf(-sh[0]);
        // loss = q_latent + 0.25*e_latent; both equal mean((q-x)^2)
        *out_loss = 1.25f * (*loss_acc) * (1.0f / 8388608.0f);
    }
}

// ---------------------------------------------------------------------------
extern "C" void kernel_launch(void* const* d_in, const int* in_sizes, int n_in,
                              void* d_out, int out_size, void* d_ws, size_t ws_size,
                              hipStream_t stream) {
    const float* x = (const float*)d_in[0];   // [32,64,64,64] NCHW fp32
    const float* w = (const float*)d_in[1];   // [512,64] fp32

    float* out      = (float*)d_out;
    float* dist     = out + OFF_DIST;
    float* quant    = out + OFF_QUANT;
    float* loss     = out + OFF_LOSS;
    float* enc      = out + OFF_ENC;
    float* enc_idx  = out + OFF_IDX;
    float* perp     = out + OFF_PERP;

    int*   counts   = (int*)d_ws;                        // 512 ints
    float* loss_acc = (float*)((char*)d_ws + 2048);      // 1 float
    float* wsq_tab  = (float*)((char*)d_ws + 2560);      // 512 floats

    vq_init_kernel<<<1, 512, 0, stream>>>(w, counts, loss_acc, wsq_tab);

    // 131072 rows / 16 rows-per-wave = 8192 waves / 8 waves-per-block
    vq_dist_kernel<<<1024, 256, 0, stream>>>(x, w, wsq_tab, dist, enc, enc_idx,
                                             counts, loss_acc);

    // 8388608 elements / 256
    vq_quant_kernel<<<32768, 256, 0, stream>>>(w, enc_idx, quant);

    vq_finalize_kernel<<<1, 512, 0, stream>>>(counts, loss_acc, loss, perp);
}